// EOPA_72129680769665
// MI455X (gfx1250) — compile-verified
//
#include <hip/hip_runtime.h>
#include <hip/hip_bf16.h>

typedef __attribute__((ext_vector_type(2))) float v2f;
typedef __attribute__((ext_vector_type(8))) float v8f;

#define BB 32
#define SS 150
#define DD 128

// ---------------------------------------------------------------------------
// Kernel 1: neigh[b,i,d] = max_j ( A[b,j,i] ? x[b,j,d] : 0 )
// One block per batch b. x[b] (150x128 f32 = 76.8KB) staged in LDS,
// A column-pair converted to float masks in LDS. 256 threads = (2 i's) x (128 d).
// ---------------------------------------------------------------------------
__global__ void EOPA_neigh_max_kernel(const float* __restrict__ x,
                                      const int* __restrict__ A,
                                      float* __restrict__ neigh) {
    extern __shared__ float smem[];
    float* xs = smem;                // SS*DD floats
    float* am = smem + SS * DD;      // 2*SS float masks for the current i-pair

    const int b   = blockIdx.x;
    const int tid = threadIdx.x;

    const float* xb = x + (size_t)b * SS * DD;
    for (int idx = tid; idx < SS * DD; idx += blockDim.x) xs[idx] = xb[idx];
    __syncthreads();

    const int* Ab = A + (size_t)b * SS * SS;
    const int c = tid >> 7;      // which i of the pair (0/1)
    const int d = tid & 127;     // feature index

    for (int i0 = 0; i0 < SS; i0 += 2) {
        // stage masks for columns (i0, i0+1): am[j*2+cc] = (float)A[b][j][i0+cc]
        for (int idx = tid; idx < SS * 2; idx += blockDim.x) {
            const int j  = idx >> 1;
            const int cc = idx & 1;
            am[idx] = (float)Ab[j * SS + i0 + cc];
        }
        __syncthreads();

        float acc = -INFINITY;
        #pragma unroll 5
        for (int j = 0; j < SS; ++j) {
            // mask in {0,1}: x*mask == where(cond, x, 0); max includes the zeros
            acc = fmaxf(acc, xs[j * DD + d] * am[j * 2 + c]);
        }
        neigh[(size_t)b * SS * DD + (size_t)(i0 + c) * DD + d] = acc;
        __syncthreads();   // am reused next iteration
    }
}

// ---------------------------------------------------------------------------
// Kernel 2: out = X @ Wself^T + NEIGH @ Wneigh^T   (M=4800, N=128, K=128)
// One wave (wave32) computes one 16x16 output tile with V_WMMA_F32_16X16X4_F32.
// 8 waves per 256-thread block; 2400 tiles -> 300 blocks.
//
// A-frag (16x4 f32, 2 VGPRs): lanes 0-15 row M=lane, K={k,k+1};
//                             lanes 16-31 row M=lane-16, K={k+2,k+3}.
// B-frag mirrored over N.  C/D: VGPR r -> (M = m0 + 8*(lane>>4) + r, N = n0 + (lane&15)).
// ---------------------------------------------------------------------------
__global__ void EOPA_gemm_wmma_kernel(const float* __restrict__ X,
                                      const float* __restrict__ NEIGH,
                                      const float* __restrict__ Wself,
                                      const float* __restrict__ Wneigh,
                                      float* __restrict__ out) {
    const int tid  = threadIdx.x;
    const int wave = tid >> 5;
    const int lane = tid & 31;

    const int tile = blockIdx.x * 8 + wave;   // 0..2399
    const int nt = tile & 7;                  // 8 N-tiles
    const int mt = tile >> 3;                 // 300 M-tiles
    const int m0 = mt * 16;
    const int n0 = nt * 16;

    const int laneLo = lane & 15;
    const int kHalf  = (lane >> 4) << 1;      // 0 or 2

    v8f acc = {};

    // ---- pass 1: X * Wself^T ----
    {
        const float* arow = X     + (size_t)(m0 + laneLo) * DD + kHalf;
        const float* brow = Wself + (size_t)(n0 + laneLo) * DD + kHalf;  // B[k][n] = W[n][k]
        #pragma unroll 8
        for (int k = 0; k < DD; k += 4) {
            v2f a = *(const v2f*)(arow + k);
            v2f b = *(const v2f*)(brow + k);
            acc = __builtin_amdgcn_wmma_f32_16x16x4_f32(
                false, a, false, b, (short)0, acc, false, false);
        }
    }
    // ---- pass 2: NEIGH * Wneigh^T, accumulated into same C ----
    {
        const float* arow = NEIGH  + (size_t)(m0 + laneLo) * DD + kHalf;
        const float* brow = Wneigh + (size_t)(n0 + laneLo) * DD + kHalf;
        #pragma unroll 8
        for (int k = 0; k < DD; k += 4) {
            v2f a = *(const v2f*)(arow + k);
            v2f b = *(const v2f*)(brow + k);
            acc = __builtin_amdgcn_wmma_f32_16x16x4_f32(
                false, a, false, b, (short)0, acc, false, false);
        }
    }

    // ---- store D ----
    const int colN    = n0 + laneLo;
    const int rowBase = m0 + ((lane >> 4) << 3);
    #pragma unroll
    for (int r = 0; r < 8; ++r) {
        out[(size_t)(rowBase + r) * DD + colN] = acc[r];
    }
}

extern "C" void kernel_launch(void* const* d_in, const int* in_sizes, int n_in,
                              void* d_out, int out_size, void* d_ws, size_t ws_size,
                              hipStream_t stream) {
    const float* x      = (const float*)d_in[0];   // (B,S,D) f32
    const int*   A      = (const int*)  d_in[1];   // (B,S,S) i32
    const float* Wself  = (const float*)d_in[2];   // (D,D)   f32
    const float* Wneigh = (const float*)d_in[3];   // (D,D)   f32
    float*       out    = (float*)d_out;           // (B,S,D) f32
    float*       neigh  = (float*)d_ws;            // (B,S,D) f32 scratch (2.4 MB)

    // Kernel 1: one block per batch; dynamic LDS = x[b] + mask pair
    const size_t lds_bytes = (size_t)(SS * DD + 2 * SS) * sizeof(float);
    EOPA_neigh_max_kernel<<<BB, 256, lds_bytes, stream>>>(x, A, neigh);

    // Kernel 2: 2400 16x16 tiles, 8 waves/block -> 300 blocks
    EOPA_gemm_wmma_kernel<<<300, 256, 0, stream>>>(x, neigh, Wself, Wneigh, out);
}